// SetAbstractionLayer_36386962932203
// MI455X (gfx1250) — compile-verified
//
#include <hip/hip_runtime.h>
#include <hip/hip_bf16.h>
#include <stdint.h>

#define B_  4
#define N_  16384
#define C_  64
#define S_  2048
#define K_  32
#define R2_ 0.04f

typedef float v2f  __attribute__((ext_vector_type(2)));
typedef float v8f  __attribute__((ext_vector_type(8)));
typedef unsigned int u32x4 __attribute__((ext_vector_type(4)));
typedef int   i32x4 __attribute__((ext_vector_type(4)));
typedef int   i32x8 __attribute__((ext_vector_type(8)));

// ------------------------------------------------------------------
// Kernel 1: farthest point sampling. One workgroup per batch; whole
// point cloud (AoS, 192KB) lives in LDS (CDNA5: 320KB/WGP). The bulk
// HBM->LDS stage is issued through the Tensor Data Mover (1D D#,
// 49152 f32 elements), then a cooperative load loop re-writes the
// same region as a correctness net. 2048 serial max-min steps with
// wave32 shfl reduction + cross-wave LDS reduction.
// ------------------------------------------------------------------
__global__ __launch_bounds__(1024) void fps_kernel(const float* __restrict__ pc,
                                                   int* __restrict__ cidx,
                                                   float* __restrict__ cc)
{
  __shared__ float sp[N_ * 3];
  __shared__ float rv[32];
  __shared__ int   ri[32];
  __shared__ int   sLast;

  const int b    = blockIdx.x;
  const int tid  = threadIdx.x;
  const int lane = tid & 31;
  const int wid  = tid >> 5;

#if __has_builtin(__builtin_amdgcn_tensor_load_to_lds)
  if (wid == 0) {
    // D# group0: count=1, lds_addr, 57-bit global_addr, type=2
    const unsigned long long ga = (unsigned long long)(pc + (size_t)b * N_ * 3);
    const unsigned lds = (unsigned)(size_t)(void*)&sp[0]; // LDS aperture low 32b == LDS offset
    u32x4 g0;
    g0.x = 1u;                                   // count=1, flags=0
    g0.y = lds;                                  // bits 63:32  lds_addr
    g0.z = (unsigned)(ga & 0xFFFFFFFFull);       // bits 95:64  global_addr[31:0]
    g0.w = (unsigned)((ga >> 32) & 0x1FFFFFFull) // bits 120:96 global_addr[56:32]
         | (2u << 30);                           // bits 127:126 type=2 ("image")
    // D# group1: data_size=4B; 1D tensor: dim0=tile0=49152, dim1=1
    i32x8 g1;
    g1[0] = (int)(2u << 16);        // data_size=2 (4B), mask/flags=0
    g1[1] = (int)0xC0000000u;       // tensor_dim0[15:0]=0xC000 in [31:16]
    g1[2] = (int)0x00010000u;       // tensor_dim0[31:16]=0, tensor_dim1[15:0]=1
    g1[3] = (int)0xC0000000u;       // tensor_dim1[31:16]=0, tile_dim0=0xC000
    g1[4] = 0;                      // tile_dim1=0 (1D), tile_dim2=0
    g1[5] = (int)49152;             // tensor_dim0_stride[31:0]
    g1[6] = 0;                      // stride0[47:32], stride1[15:0]
    g1[7] = 0;                      // stride1[47:16]
    i32x4 g2 = {0, 0, 0, 0};
    i32x4 g3 = {0, 0, 0, 0};
    i32x8 g4 = {0, 0, 0, 0, 0, 0, 0, 0};  // clang-23 6-arg form: extra group, zero-filled
    __builtin_amdgcn_tensor_load_to_lds(g0, g1, g2, g3, g4, 0);
    __builtin_amdgcn_s_wait_tensorcnt(0);
  }
  __syncthreads();
#endif
  // Correctness net: rewrite the same AoS data with plain loads.
  for (int i = tid; i < N_; i += 1024) {
    const float* p = pc + ((size_t)b * N_ + i) * 3;
    sp[3 * i]     = p[0];
    sp[3 * i + 1] = p[1];
    sp[3 * i + 2] = p[2];
  }
  __syncthreads();

  float dist[16];
#pragma unroll
  for (int j = 0; j < 16; ++j) dist[j] = __builtin_inff();

  int last = 0;
  for (int s = 0; s < S_; ++s) {
    if (tid == 0) {                       // lax.scan emits the carry-in `last`
      cidx[b * S_ + s] = last;
      float* c = cc + ((size_t)b * S_ + s) * 3;
      c[0] = sp[3 * last]; c[1] = sp[3 * last + 1]; c[2] = sp[3 * last + 2];
    }
    const float px = sp[3 * last], py = sp[3 * last + 1], pz = sp[3 * last + 2];
    float bv = -1.0f; int bi = 0x7FFFFFFF;
#pragma unroll
    for (int j = 0; j < 16; ++j) {
      const int i = j * 1024 + tid;       // dword stride 3 across lanes: gcd(3,64)=1, no conflicts
      const float dx = sp[3 * i] - px, dy = sp[3 * i + 1] - py, dz = sp[3 * i + 2] - pz;
      const float d  = dx * dx + dy * dy + dz * dz;
      const float nd = fminf(dist[j], d);
      dist[j] = nd;
      if (nd > bv || (nd == bv && i < bi)) { bv = nd; bi = i; }
    }
    // wave32 argmax (ties -> smallest index, matching jnp.argmax)
#pragma unroll
    for (int off = 16; off > 0; off >>= 1) {
      const float ov = __shfl_xor(bv, off, 32);
      const int   oi = __shfl_xor(bi, off, 32);
      if (ov > bv || (ov == bv && oi < bi)) { bv = ov; bi = oi; }
    }
    if (lane == 0) { rv[wid] = bv; ri[wid] = bi; }
    __syncthreads();
    if (wid == 0) {
      float v2 = rv[lane]; int i2 = ri[lane];
#pragma unroll
      for (int off = 16; off > 0; off >>= 1) {
        const float ov = __shfl_xor(v2, off, 32);
        const int   oi = __shfl_xor(i2, off, 32);
        if (ov > v2 || (ov == v2 && oi < i2)) { v2 = ov; i2 = oi; }
      }
      if (lane == 0) sLast = i2;
    }
    __syncthreads();
    last = sLast;
  }
}

// ------------------------------------------------------------------
// Kernel 2: ball query via V_WMMA_F32_16X16X4_F32, 32 columns per
// iteration (2 WMMAs) with software-pipelined coordinate loads.
// A[m] = (-2cx,-2cy,-2cz, c2), B[n] = (px,py,pz, 1), Cacc = x2[n]
//   => D[m][n] = squared distance directly.
// D layout: vgpr v, lanes 0-15 -> row v, lanes 16-31 -> row v+8,
// col = lane%16 => ballot bit == column. Two 16-bit row masks concat
// into one 32-bit mask; lane r<16 appends ascending indices (exactly
// the reference's "first K in index order"), uniform early exit.
// ------------------------------------------------------------------
__global__ __launch_bounds__(256) void ballq_kernel(const float* __restrict__ pc,
                                                    const float* __restrict__ cc,
                                                    int* __restrict__ gidx)
{
  const int  lane = threadIdx.x & 31;
  const int  tile = blockIdx.x * 8 + (int)(threadIdx.x >> 5); // 16 centroids per wave
  const int  b    = tile >> 7;                                // 128 tiles per batch
  const int  trow = tile & 127;
  const int  m    = lane & 15;
  const bool hi   = lane >= 16;

  const int    row0 = trow * 16;
  const float* c    = cc + ((size_t)b * S_ + row0 + m) * 3;
  const float  cx = c[0], cy = c[1], cz = c[2];
  const float  c2 = cx * cx + cy * cy + cz * cz;
  v2f a;                               // 16x4 f32 A: vgpr0=K0/K2, vgpr1=K1/K3
  a.x = hi ? (-2.0f * cz) : (-2.0f * cx);
  a.y = hi ? c2           : (-2.0f * cy);

  int          cnt      = hi ? K_ : 0; // upper-half lanes never append
  int          firstIdx = 0;
  const size_t rowBase  = ((size_t)b * S_ + row0 + m) * K_;
  const float* pb       = pc + (size_t)b * N_ * 3;

  // prologue loads: columns [0,16) and [16,32), point index = chunk + m
  float px0 = pb[(size_t)m * 3], py0 = pb[(size_t)m * 3 + 1], pz0 = pb[(size_t)m * 3 + 2];
  float px1 = pb[(size_t)(16 + m) * 3], py1 = pb[(size_t)(16 + m) * 3 + 1],
        pz1 = pb[(size_t)(16 + m) * 3 + 2];

  for (int n0 = 0; n0 < N_; n0 += 32) {
    // issue next iteration's loads early (clamped: branch-free)
    const int nnext = (n0 + 32 < N_) ? (n0 + 32) : (N_ - 32);
    const float* q0 = pb + (size_t)(nnext + m) * 3;
    const float* q1 = pb + (size_t)(nnext + 16 + m) * 3;
    const float nx0 = q0[0], ny0 = q0[1], nz0 = q0[2];
    const float nx1 = q1[0], ny1 = q1[1], nz1 = q1[2];

    const float x20 = px0 * px0 + py0 * py0 + pz0 * pz0;
    const float x21 = px1 * px1 + py1 * py1 + pz1 * pz1;
    v2f bf0, bf1;
    bf0.x = hi ? pz0 : px0;  bf0.y = hi ? 1.0f : py0;
    bf1.x = hi ? pz1 : px1;  bf1.y = hi ? 1.0f : py1;
    v8f acc0, acc1;
#pragma unroll
    for (int v = 0; v < 8; ++v) { acc0[v] = x20; acc1[v] = x21; }

    v8f d0 = __builtin_amdgcn_wmma_f32_16x16x4_f32(false, a, false, bf0, (short)0, acc0, false, false);
    v8f d1 = __builtin_amdgcn_wmma_f32_16x16x4_f32(false, a, false, bf1, (short)0, acc1, false, false);

    const unsigned a0 = __builtin_amdgcn_ballot_w32(d0[0] < R2_);
    const unsigned a1 = __builtin_amdgcn_ballot_w32(d0[1] < R2_);
    const unsigned a2 = __builtin_amdgcn_ballot_w32(d0[2] < R2_);
    const unsigned a3 = __builtin_amdgcn_ballot_w32(d0[3] < R2_);
    const unsigned a4 = __builtin_amdgcn_ballot_w32(d0[4] < R2_);
    const unsigned a5 = __builtin_amdgcn_ballot_w32(d0[5] < R2_);
    const unsigned a6 = __builtin_amdgcn_ballot_w32(d0[6] < R2_);
    const unsigned a7 = __builtin_amdgcn_ballot_w32(d0[7] < R2_);
    const unsigned b0 = __builtin_amdgcn_ballot_w32(d1[0] < R2_);
    const unsigned b1 = __builtin_amdgcn_ballot_w32(d1[1] < R2_);
    const unsigned b2 = __builtin_amdgcn_ballot_w32(d1[2] < R2_);
    const unsigned b3 = __builtin_amdgcn_ballot_w32(d1[3] < R2_);
    const unsigned b4 = __builtin_amdgcn_ballot_w32(d1[4] < R2_);
    const unsigned b5 = __builtin_amdgcn_ballot_w32(d1[5] < R2_);
    const unsigned b6 = __builtin_amdgcn_ballot_w32(d1[6] < R2_);
    const unsigned b7 = __builtin_amdgcn_ballot_w32(d1[7] < R2_);

    // row r: word = x[r&7]; rows 0-7 in low16, rows 8-15 in high16
    const unsigned wA =
        (m & 4) ? ((m & 2) ? ((m & 1) ? a7 : a6) : ((m & 1) ? a5 : a4))
                : ((m & 2) ? ((m & 1) ? a3 : a2) : ((m & 1) ? a1 : a0));
    const unsigned wB =
        (m & 4) ? ((m & 2) ? ((m & 1) ? b7 : b6) : ((m & 1) ? b5 : b4))
                : ((m & 2) ? ((m & 1) ? b3 : b2) : ((m & 1) ? b1 : b0));
    const unsigned mA = (m < 8) ? (wA & 0xFFFFu) : (wA >> 16);
    const unsigned mB = (m < 8) ? (wB & 0xFFFFu) : (wB >> 16);
    unsigned mask32 = mA | (mB << 16);   // bit j == column n0+j

    while (mask32 && cnt < K_) {
      const int nn = __builtin_ctz(mask32);
      mask32 &= mask32 - 1u;
      const int gi = n0 + nn;
      if (cnt == 0) firstIdx = gi;
      gidx[rowBase + cnt] = gi;
      ++cnt;
    }
    if (__builtin_amdgcn_ballot_w32(cnt < K_) == 0u) break; // uniform early exit

    px0 = nx0; py0 = ny0; pz0 = nz0;   // rotate pipeline
    px1 = nx1; py1 = ny1; pz1 = nz1;
  }

  if (!hi) { // pad with first hit (CUDA ball_query semantics); 0 if empty
    const int pad = (cnt == 0) ? 0 : firstIdx;
    for (int k = cnt; k < K_; ++k) gidx[rowBase + k] = pad;
  }
}

// ------------------------------------------------------------------
// Kernel 3: gather + concat -> (B, 3+C, S, K). One thread per (b,s,k);
// k fastest => coalesced 128B stores per channel plane; feature gathers
// hit a 64KB L2-resident window per (b,c).
// ------------------------------------------------------------------
__global__ __launch_bounds__(256) void group_kernel(const float* __restrict__ pc,
                                                    const float* __restrict__ feat,
                                                    const float* __restrict__ cc,
                                                    const int* __restrict__ gidx,
                                                    float* __restrict__ out)
{
  const int t   = blockIdx.x * 256 + (int)threadIdx.x; // B*S*K threads
  const int k   = t & (K_ - 1);
  const int s   = (t >> 5) & (S_ - 1);
  const int b   = t >> 16;                              // S*K == 65536
  const int idx = gidx[t];

  const float* p = pc + ((size_t)b * N_ + idx) * 3;
  const float* c = cc + ((size_t)b * S_ + s) * 3;
  const size_t plane = (size_t)S_ * K_;
  const size_t ob    = (size_t)b * (3 + C_) * plane + (size_t)s * K_ + k;

  out[ob]             = p[0] - c[0];
  out[ob + plane]     = p[1] - c[1];
  out[ob + 2 * plane] = p[2] - c[2];

  const float* fb = feat + (size_t)b * C_ * N_ + idx;
#pragma unroll 4
  for (int ch = 0; ch < C_; ++ch)
    out[ob + (size_t)(3 + ch) * plane] = fb[(size_t)ch * N_];
}

extern "C" void kernel_launch(void* const* d_in, const int* in_sizes, int n_in,
                              void* d_out, int out_size, void* d_ws, size_t ws_size,
                              hipStream_t stream)
{
  (void)in_sizes; (void)n_in; (void)out_size; (void)ws_size;
  const float* pc   = (const float*)d_in[0];   // (B,N,3) f32
  const float* feat = (const float*)d_in[1];   // (B,C,N) f32

  char*  ws   = (char*)d_ws;
  int*   cidx = (int*)ws;                                          // B*S      (32KB)
  float* cc   = (float*)(ws + (size_t)B_ * S_ * sizeof(int));      // B*S*3    (96KB)
  int*   gidx = (int*)(ws + (size_t)B_ * S_ * sizeof(int)
                          + (size_t)B_ * S_ * 3 * sizeof(float));  // B*S*K    (1MB)
  float* out  = (float*)d_out;

  fps_kernel  <<<B_, 1024, 0, stream>>>(pc, cidx, cc);
  ballq_kernel<<<(B_ * S_ / 16) / 8, 256, 0, stream>>>(pc, cc, gidx);
  group_kernel<<<(B_ * S_ * K_) / 256, 256, 0, stream>>>(pc, feat, cc, gidx, out);
}